// WassersteinLoss_81767587381854
// MI455X (gfx1250) — compile-verified
//
#include <hip/hip_runtime.h>
#include <hip/hip_bf16.h>

// ---------------------------------------------------------------------------
// Wasserstein-style loss, MI455X (gfx1250, wave32).
// Memory-bound (64MB streamed once, ~2.9us floor at 23.3TB/s).
// Tridiagonal spline solve recast as f32 WMMA GEMM against precomputed A^-1.
// ---------------------------------------------------------------------------

#define NROW 2048       // NS*NR
#define NT   4096
#define NP   100
#define PAD  112        // 98 padded to multiple of 16
#define NI   98         // interior unknowns of natural cubic spline

typedef float v2f __attribute__((ext_vector_type(2)));
typedef float v8f __attribute__((ext_vector_type(8)));

// ---------------------------------------------------------------------------
// k_init: A = tri(1,4,1), 98x98.  Solve A x = e_j for j=0..97 (Thomas) to get
// A^-1 columns; store zero-padded 112x112 (A symmetric -> orientation free).
// ---------------------------------------------------------------------------
__global__ void k_init(float* __restrict__ Ainv) {
  __shared__ float cp[NI];          // c'_i = 1/(4 - c'_{i-1})
  __shared__ float dp[NI * NI];     // forward-pass d' per column
  const int tid = threadIdx.x;

  for (int i = tid; i < PAD * PAD; i += blockDim.x) Ainv[i] = 0.0f;

  if (tid == 0) {
    float c = 0.0f;
    for (int i = 0; i < NI; ++i) { c = 1.0f / (4.0f - c); cp[i] = c; }
  }
  __syncthreads();

  if (tid < NI) {
    const int j = tid;
    float prev = 0.0f;
    for (int i = 0; i < NI; ++i) {
      float e = (i == j) ? 1.0f : 0.0f;
      prev = (e - prev) * cp[i];          // cp[i] == 1/(4 - cp[i-1])
      dp[i * NI + j] = prev;
    }
    float x = dp[(NI - 1) * NI + j];
    Ainv[(NI - 1) * PAD + j] = x;
    for (int i = NI - 2; i >= 0; --i) {
      x = dp[i * NI + j] - cp[i] * x;
      Ainv[i * PAD + j] = x;
    }
  }
}

// ---------------------------------------------------------------------------
// k_obs: per row -> CDF of |obs| in LDS, inverse-CDF sample at 100 points,
// emit Q row and spline RHS row (zero-padded to 112).
// ---------------------------------------------------------------------------
__global__ void __launch_bounds__(256)
k_obs(const float* __restrict__ obs, const float* __restrict__ t,
      float* __restrict__ Qb, float* __restrict__ RHSb) {
  __shared__ float Fv[NT];
  __shared__ float part[256];
  __shared__ float Qs[NP];
  __shared__ float sc[2];            // y0, yLast

  const int row = blockIdx.x;
  const int tid = threadIdx.x;
  const float dtv = t[1] - t[0];

  const float4* rp = reinterpret_cast<const float4*>(obs + (size_t)row * NT);

  float v[16], pre[16];
  float run = 0.0f;
#pragma unroll
  for (int q = 0; q < 4; ++q) {
    float4 x = rp[tid * 4 + q];
    float a0 = fabsf(x.x), a1 = fabsf(x.y), a2 = fabsf(x.z), a3 = fabsf(x.w);
    v[q*4+0]=a0; run+=a0; pre[q*4+0]=run;
    v[q*4+1]=a1; run+=a1; pre[q*4+1]=run;
    v[q*4+2]=a2; run+=a2; pre[q*4+2]=run;
    v[q*4+3]=a3; run+=a3; pre[q*4+3]=run;
  }
  if (tid == 0)   sc[0] = v[0];
  if (tid == 255) sc[1] = v[15];
  part[tid] = run;
  __syncthreads();
  // Hillis-Steele inclusive scan over 256 partials
  for (int off = 1; off < 256; off <<= 1) {
    float add = (tid >= off) ? part[tid - off] : 0.0f;
    __syncthreads();
    part[tid] += add;
    __syncthreads();
  }
  const float total = part[255];
  const float base  = part[tid] - run;         // exclusive prefix
  const float y0 = sc[0], yl = sc[1];
  const float norm = dtv * (total - 0.5f * (y0 + yl));
  const float invn = 1.0f / norm;

#pragma unroll
  for (int k = 0; k < 16; ++k) {
    float S = base + pre[k];
    Fv[tid * 16 + k] = (S - 0.5f * (y0 + v[k])) * dtv * invn;
  }
  __syncthreads();

  // inverse CDF: Q[k] = interp(p_k, Fv, t)
  if (tid < NP) {
    float p = (float)tid / 99.0f;
    float q;
    if (p <= Fv[0])            q = t[0];
    else if (p >= Fv[NT - 1])  q = t[NT - 1];
    else {
      int lo = 0, hi = NT - 1;
      while (hi - lo > 1) {
        int mid = (lo + hi) >> 1;
        if (Fv[mid] <= p) lo = mid; else hi = mid;
      }
      float f0 = Fv[lo], f1 = Fv[hi];
      q = t[lo] + (p - f0) * (t[hi] - t[lo]) / (f1 - f0);
    }
    Qs[tid] = q;
    Qb[(size_t)row * PAD + tid] = q;
  }
  __syncthreads();

  if (tid < PAD) {
    const float h = 1.0f / 99.0f;
    float r = 0.0f;
    if (tid < NI)
      r = 6.0f * (Qs[tid + 2] - 2.0f * Qs[tid + 1] + Qs[tid]) / (h * h);
    RHSb[(size_t)row * PAD + tid] = r;
  }
}

// ---------------------------------------------------------------------------
// k_gemm: M(2048x112) = RHS(2048x112) @ Ainv(112x112), f32 WMMA 16x16x4.
// One wave per 16x16 output tile; 28 chained k=4 WMMA steps.
// Fragment layout per ISA 7.12.2:
//   A 16x4 : lanes 0-15 (M=lane) hold K=0/1 in v0/v1; lanes 16-31 K=2/3.
//   B 4x16 : lanes 0-15 (N=lane) hold K=0/1; lanes 16-31 K=2/3.
//   D 16x16: VGPR g -> row g (lanes 0-15) / row g+8 (lanes 16-31), N=lane%16.
// ---------------------------------------------------------------------------
__global__ void __launch_bounds__(32)
k_gemm(const float* __restrict__ RHSb, const float* __restrict__ Ainv,
       float* __restrict__ Mb) {
  const int tileN = blockIdx.x;      // 0..6
  const int tileM = blockIdx.y;      // 0..127
  const int lane  = threadIdx.x;     // 0..31
  const int half  = (lane >= 16) ? 1 : 0;
  const int l     = lane & 15;

  const float* Arow = RHSb + (size_t)(tileM * 16 + l) * PAD;
  const int colB = tileN * 16 + l;

  v8f acc = {};
  for (int k = 0; k < PAD; k += 4) {
    v2f a, b;
    a.x = Arow[k + (half ? 2 : 0)];
    a.y = Arow[k + (half ? 3 : 1)];
    b.x = Ainv[(k + (half ? 2 : 0)) * PAD + colB];
    b.y = Ainv[(k + (half ? 3 : 1)) * PAD + colB];
    acc = __builtin_amdgcn_wmma_f32_16x16x4_f32(
        /*neg_a=*/false, a, /*neg_b=*/false, b,
        /*c_mod=*/(short)0, acc, /*reuse_a=*/false, /*reuse_b=*/false);
  }
#pragma unroll
  for (int g = 0; g < 8; ++g) {
    int r = tileM * 16 + g + (half ? 8 : 0);
    Mb[(size_t)r * PAD + tileN * 16 + l] = acc[g];
  }
}

// ---------------------------------------------------------------------------
// k_f: per row -> spline coeffs in LDS, stream |f|, scan -> CDF, spline eval,
// trapezoid-weighted accumulate -> per-row partial (no atomics).
// ---------------------------------------------------------------------------
__global__ void __launch_bounds__(256)
k_f(const float* __restrict__ f, const float* __restrict__ t,
    const float* __restrict__ Qb, const float* __restrict__ Mb,
    float* __restrict__ partials) {
  __shared__ float part[256];
  __shared__ float Qs[NP], Mf[NP];
  __shared__ float ca[99], cb[99], cc[99], cd[99];
  __shared__ float sc[2];

  const int row = blockIdx.x;
  const int tid = threadIdx.x;
  const float dtv = t[1] - t[0];
  const float h = 1.0f / 99.0f;
  const float xmax = 99.0f * h;

  const float4* rp = reinterpret_cast<const float4*>(f + (size_t)row * NT);
  const float4* tp = reinterpret_cast<const float4*>(t);

  // warm the streaming read while we set up spline coeffs (global_prefetch_b8)
  __builtin_prefetch(rp + tid * 4, 0, 0);

  if (tid < NP) {
    Qs[tid] = Qb[(size_t)row * PAD + tid];
    Mf[tid] = (tid == 0 || tid == NP - 1) ? 0.0f
                                          : Mb[(size_t)row * PAD + (tid - 1)];
  }
  __syncthreads();
  if (tid < 99) {
    ca[tid] = Qs[tid];
    cb[tid] = (Qs[tid + 1] - Qs[tid]) / h
            - h * (2.0f * Mf[tid] + Mf[tid + 1]) / 6.0f;
    cc[tid] = Mf[tid] * 0.5f;
    cd[tid] = (Mf[tid + 1] - Mf[tid]) / (6.0f * h);
  }

  float v[16], pre[16];
  float run = 0.0f;
#pragma unroll
  for (int q = 0; q < 4; ++q) {
    float4 x = rp[tid * 4 + q];
    float a0 = fabsf(x.x), a1 = fabsf(x.y), a2 = fabsf(x.z), a3 = fabsf(x.w);
    v[q*4+0]=a0; run+=a0; pre[q*4+0]=run;
    v[q*4+1]=a1; run+=a1; pre[q*4+1]=run;
    v[q*4+2]=a2; run+=a2; pre[q*4+2]=run;
    v[q*4+3]=a3; run+=a3; pre[q*4+3]=run;
  }
  if (tid == 0)   sc[0] = v[0];
  if (tid == 255) sc[1] = v[15];
  part[tid] = run;
  __syncthreads();
  for (int off = 1; off < 256; off <<= 1) {
    float add = (tid >= off) ? part[tid - off] : 0.0f;
    __syncthreads();
    part[tid] += add;
    __syncthreads();
  }
  const float total = part[255];
  const float base  = part[tid] - run;
  const float y0 = sc[0], yl = sc[1];
  const float norm = dtv * (total - 0.5f * (y0 + yl));
  const float invn = 1.0f / norm;

  float accum = 0.0f;
#pragma unroll
  for (int q = 0; q < 4; ++q) {
    float4 tv = tp[tid * 4 + q];
    float tvals[4] = {tv.x, tv.y, tv.z, tv.w};
#pragma unroll
    for (int j = 0; j < 4; ++j) {
      int k = q * 4 + j;
      int gi = tid * 16 + k;
      float S  = base + pre[k];
      float Fi = (S - 0.5f * (y0 + v[k])) * dtv * invn;   // CDF of f
      float ft = v[k] * invn;                              // f_tilde
      float xi = fminf(fmaxf(Fi, 0.0f), xmax);
      int idx = (int)(xi / h);
      idx = min(max(idx, 0), 98);
      float dxv = xi - (float)idx * h;
      float s = ca[idx] + dxv * (cb[idx] + dxv * (cc[idx] + dxv * cd[idx]));
      float diff = tvals[j] - s;
      float w = (gi == 0 || gi == NT - 1) ? 0.5f : 1.0f;
      accum += w * diff * diff * ft;
    }
  }
  part[tid] = accum;
  __syncthreads();
  for (int s = 128; s > 0; s >>= 1) {
    if (tid < s) part[tid] += part[tid + s];
    __syncthreads();
  }
  if (tid == 0) partials[row] = dtv * part[0];
}

// ---------------------------------------------------------------------------
// k_reduce: deterministic fixed-tree sum of 2048 row partials.
// ---------------------------------------------------------------------------
__global__ void __launch_bounds__(256)
k_reduce(const float* __restrict__ partials, float* __restrict__ out) {
  __shared__ float sm[256];
  const int tid = threadIdx.x;
  float a = 0.0f;
  for (int i = tid; i < NROW; i += 256) a += partials[i];
  sm[tid] = a;
  __syncthreads();
  for (int s = 128; s > 0; s >>= 1) {
    if (tid < s) sm[tid] += sm[tid + s];
    __syncthreads();
  }
  if (tid == 0) out[0] = sm[0];
}

// ---------------------------------------------------------------------------
extern "C" void kernel_launch(void* const* d_in, const int* in_sizes, int n_in,
                              void* d_out, int out_size, void* d_ws, size_t ws_size,
                              hipStream_t stream) {
  const float* f   = (const float*)d_in[0];
  const float* obs = (const float*)d_in[1];
  const float* t   = (const float*)d_in[2];
  float* out = (float*)d_out;

  float* ws       = (float*)d_ws;
  float* Ainv     = ws;                              // 112*112      = 12544
  float* Qb       = Ainv + PAD * PAD;                // 2048*112     = 229376
  float* RHSb     = Qb   + (size_t)NROW * PAD;       // 2048*112
  float* Mb       = RHSb + (size_t)NROW * PAD;       // 2048*112
  float* partials = Mb   + (size_t)NROW * PAD;       // 2048

  k_init<<<1, 128, 0, stream>>>(Ainv);
  k_obs<<<NROW, 256, 0, stream>>>(obs, t, Qb, RHSb);
  k_gemm<<<dim3(PAD / 16, NROW / 16), 32, 0, stream>>>(RHSb, Ainv, Mb);
  k_f<<<NROW, 256, 0, stream>>>(f, t, Qb, Mb, partials);
  k_reduce<<<1, 256, 0, stream>>>(partials, out);
}